// ConvCapsuleLayer3D_8538394984751
// MI455X (gfx1250) — compile-verified
//
#include <hip/hip_runtime.h>

typedef __attribute__((ext_vector_type(2))) float v2f;
typedef __attribute__((ext_vector_type(8))) float v8f;

#define SP   32
#define SP2  1024
#define SP3  32768
#define ICAP 4
#define IA   16
#define NC   8
#define NA   32
#define NCH  256      // NC*NA
#define KTOT 432      // IA*27
#define CIS  18       // ci stride in LDS (even: 8B alignment; conflict-free)

// ---------------------------------------------------------------------------
// Kernel 0: pack W (O=256, I=16, 3,3,3) into K-pair-interleaved layout:
//   kk = r*16 + ci  (r = kd*9+kh*3+kw spatial-major, ci minor)
//   Wq[(kk>>1)*512 + n*2 + (kk&1)]
// so a B fragment (rows kk, kk+1 at column n) is one aligned 8-byte load.
// ---------------------------------------------------------------------------
__global__ void pack_w(const float* __restrict__ W, float* __restrict__ Wq) {
    int idx = blockIdx.x * blockDim.x + threadIdx.x;
    if (idx >= NCH * KTOT) return;
    int n   = idx / KTOT;
    int rem = idx % KTOT;
    int ci  = rem / 27;
    int r   = rem % 27;
    int kk  = r * 16 + ci;
    Wq[((size_t)(kk >> 1) * NCH + n) * 2 + (kk & 1)] = W[idx];
}

// ---------------------------------------------------------------------------
// Kernel 1: implicit-GEMM 3D conv via V_WMMA_F32_16X16X4_F32.
// Block = 256 threads = 8 waves, one block per (img, d, h) row.
// x tile (3x3 d/h halo, zero-padded w halo) staged once in LDS;
// inner loop: 1 ds_load_b64 (A) + 4 global_load_b64 (B) + 4 WMMA per K-step.
// Votes stored as V[img][p][ch], ch = c*32 + a.
// ---------------------------------------------------------------------------
__global__ __launch_bounds__(256) void conv_votes(
    const float* __restrict__ X, const float* __restrict__ Wq,
    float* __restrict__ V) {
  __shared__ float xs[9 * 34 * CIS];     // [(dl*3+hl)*34 + (w+1)][ci]

  int nimg = blockIdx.x >> 10;
  int d    = (blockIdx.x >> 5) & 31;
  int h    = blockIdx.x & 31;
  int tid  = threadIdx.x;

  const float* xbase = X + (size_t)nimg * IA * SP3;

  // ---- stage x tile with zero halo (boundary logic paid once per block) ----
  for (int e = tid; e < 16 * 9 * 34; e += 256) {
    int wi = e % 34;
    int t2 = e / 34;
    int ph = t2 % 9;                     // dl*3 + hl
    int ci = t2 / 9;
    int dp = d + (ph / 3) - 1;
    int hp = h + (ph % 3) - 1;
    int wp = wi - 1;
    bool ok = (unsigned)dp < 32u && (unsigned)hp < 32u && (unsigned)wp < 32u;
    int dpc = dp < 0 ? 0 : (dp > 31 ? 31 : dp);
    int hpc = hp < 0 ? 0 : (hp > 31 ? 31 : hp);
    int wpc = wp < 0 ? 0 : (wp > 31 ? 31 : wp);
    float v = xbase[(size_t)ci * SP3 + dpc * SP2 + hpc * SP + wpc];
    xs[(ph * 34 + wi) * CIS + ci] = ok ? v : 0.0f;
  }
  __syncthreads();

  int lane = tid & 31;
  int wave = tid >> 5;
  int w0   = (wave & 1) * 16;            // M tile: 16 consecutive w
  int n0   = (wave >> 1) * 64;           // N group: 64 channels
  int half = lane >> 4;                  // K rows {0,1} vs {2,3}
  int m    = lane & 15;                  // M row / N column

  v8f acc[4];
#pragma unroll
  for (int t = 0; t < 4; ++t) acc[t] = (v8f){};

  // single B base; all loads below use compile-time immediate offsets
  const float* wqb = Wq + ((size_t)half * NCH + n0 + m) * 2;

  for (int kd = 0; kd < 3; ++kd) {
    int dp = d + kd - 1;
    if ((unsigned)dp >= 32u) continue;                 // uniform scalar skip
    for (int kh = 0; kh < 3; ++kh) {
      int hp = h + kh - 1;
      if ((unsigned)hp >= 32u) continue;               // uniform scalar skip
      const float* xsrow =
          xs + (((kd * 3 + kh) * 34) + w0 + m) * CIS + half * 2;
#pragma unroll
      for (int kw = 0; kw < 3; ++kw) {
        int r = kd * 9 + kh * 3 + kw;
        const float* wr = wqb + (size_t)r * 8 * 512;   // r*8 k-pairs
#pragma unroll
        for (int ci0 = 0; ci0 < 16; ci0 += 4) {        // K-step of 4
          v2f A = *(const v2f*)(xsrow + kw * CIS + ci0);   // ds_load_b64
#pragma unroll
          for (int t = 0; t < 4; ++t) {
            v2f Bv = *(const v2f*)(wr + (ci0 >> 1) * 512 + t * 32);
            acc[t] = __builtin_amdgcn_wmma_f32_16x16x4_f32(
                false, A, false, Bv, (short)0, acc[t], false, false);
          }
        }
      }
    }
  }

  // ---- store: C/D layout M = vgpr + 8*half, N = lane&15 ----
  float* vout = V + ((size_t)nimg * SP3 + (size_t)d * SP2 + h * SP) * NCH;
#pragma unroll
  for (int t = 0; t < 4; ++t) {
#pragma unroll
    for (int rr = 0; rr < 8; ++rr) {
      int wpos = w0 + rr + half * 8;
      vout[(size_t)wpos * NCH + (n0 + t * 16 + m)] = acc[t][rr];
    }
  }
}

// ---------------------------------------------------------------------------
// Kernel 2: fused 3-iteration sigmoid routing + squash.
// Block = 1024 threads = 32 waves; one block per (b, c, 32-p tile).
// lane = atom a (NA == wave32); votes live in registers across all
// iterations; squash/agreement via wave shuffle reductions; output
// transposed through LDS for coalesced spatial stores.
// ---------------------------------------------------------------------------
__global__ __launch_bounds__(1024) void routing(
    const float* __restrict__ V, const float* __restrict__ bias,
    float* __restrict__ out) {
  int bc    = blockIdx.x & 15;
  int ptile = blockIdx.x >> 4;
  int b = bc >> 3;
  int c = bc & 7;
  int tid  = threadIdx.x;
  int wave = tid >> 5;
  int a    = tid & 31;
  int p    = ptile * 32 + wave;

  __shared__ float sact[32][33];

  size_t chix = (size_t)c * NA + a;
  float v0 = V[((size_t)(b * ICAP + 0) * SP3 + p) * NCH + chix];
  float v1 = V[((size_t)(b * ICAP + 1) * SP3 + p) * NCH + chix];
  float v2 = V[((size_t)(b * ICAP + 2) * SP3 + p) * NCH + chix];
  float v3 = V[((size_t)(b * ICAP + 3) * SP3 + p) * NCH + chix];
  float bb = bias[c * NA + a];

  float l0 = 1.f, l1 = 1.f, l2 = 1.f, l3 = 1.f;
  float act = 0.f;
#pragma unroll
  for (int it = 0; it < 3; ++it) {
    float r0 = 1.f / (1.f + __expf(-l0));
    float r1 = 1.f / (1.f + __expf(-l1));
    float r2 = 1.f / (1.f + __expf(-l2));
    float r3 = 1.f / (1.f + __expf(-l3));
    float pre = bb + r0 * v0 + r1 * v1 + r2 * v2 + r3 * v3;
    float sq = pre * pre;
#pragma unroll
    for (int off = 16; off; off >>= 1) sq += __shfl_xor(sq, off, 32);
    act = (sq / (1.f + sq)) * pre / sqrtf(sq + 1e-9f);
    if (it < 2) {
      float d0 = v0 * act, d1 = v1 * act, d2 = v2 * act, d3 = v3 * act;
#pragma unroll
      for (int off = 16; off; off >>= 1) {
        d0 += __shfl_xor(d0, off, 32);
        d1 += __shfl_xor(d1, off, 32);
        d2 += __shfl_xor(d2, off, 32);
        d3 += __shfl_xor(d3, off, 32);
      }
      l0 += d0; l1 += d1; l2 += d2; l3 += d3;
    }
  }

  sact[wave][a] = act;          // [p_local][atom]
  __syncthreads();
  int aw = tid >> 5;            // atom
  int pl = tid & 31;            // p_local -> consecutive lanes, coalesced
  out[((size_t)(b * NC + c) * NA + aw) * SP3 + (size_t)ptile * 32 + pl] =
      sact[pl][aw];
}

// ---------------------------------------------------------------------------
extern "C" void kernel_launch(void* const* d_in, const int* in_sizes, int n_in,
                              void* d_out, int out_size, void* d_ws, size_t ws_size,
                              hipStream_t stream) {
  (void)in_sizes; (void)n_in; (void)out_size; (void)ws_size;
  const float* x    = (const float*)d_in[0];   // (2,4,16,32,32,32)
  const float* W    = (const float*)d_in[1];   // (256,16,3,3,3)
  const float* bias = (const float*)d_in[2];   // (8,32,1,1,1) -> 256
  float* out = (float*)d_out;                  // (2,8,32,32,32,32)

  char*  ws = (char*)d_ws;
  float* Wq = (float*)ws;                      // 432*256*4   = 442 KB
  float* V  = (float*)(ws + (512u << 10));     // votes: 8*32768*256*4 = 268 MB

  pack_w<<<(NCH * KTOT + 255) / 256, 256, 0, stream>>>(W, Wq);
  conv_votes<<<8 * 32 * 32, 256, 0, stream>>>(x, Wq, V);
  routing<<<16 * 1024, 1024, 0, stream>>>(V, bias, out);
}